// GetLaplacian_66305705115958
// MI455X (gfx1250) — compile-verified
//
#include <hip/hip_runtime.h>

typedef __attribute__((ext_vector_type(2))) float v2f;
typedef __attribute__((ext_vector_type(4))) float v4f;
typedef __attribute__((ext_vector_type(8))) float v8f;

#define N_DIM 2048
#define B_DIM 16
#define COL_CHUNKS 8                       // 2048 cols / 8 = 256 cols per chunk
#define BN (B_DIM * N_DIM)                 // 32768 rows total

// ---------------------------------------------------------------------------
// Pass 1a: partial row sums via V_WMMA_F32_16X16X4_F32 (B = ones(4x16)).
// Each wave owns 16 rows x 256 columns (one column chunk): 64 K=4 wmma steps.
// 16384 waves total -> ~32 MB of loads in flight device-wide (vs 18 MB
// needed at 23.3 TB/s x ~800 ns), so the pass can run at the HBM roofline.
// A-operand fp32 16x4 layout (ISA 7.12.2): lane l loads float2 at
// row (l%16), col 4*s + 2*(l/16).
// C/D layout: VGPR r: lanes 0-15 -> M=r, lanes 16-31 -> M=8+r.
// ---------------------------------------------------------------------------
__global__ __launch_bounds__(256)
void laplacian_degree_partial(const float* __restrict__ A,
                              float* __restrict__ partial) {
    const int wave = threadIdx.x >> 5;
    const int lane = threadIdx.x & 31;

    const int strips_per_batch = N_DIM / 128;              // 8 waves * 16 rows
    const int chunk  = blockIdx.x % COL_CHUNKS;
    const int strip  = blockIdx.x / COL_CHUNKS;
    const int batch     = strip / strips_per_batch;
    const int block_row = (strip % strips_per_batch) * 128;
    const int row_base  = block_row + wave * 16;

    const int row = row_base + (lane & 15);
    const int khi = lane >> 4;                             // 0 or 1
    const float* p = A + ((size_t)batch * N_DIM + row) * N_DIM
                       + chunk * (N_DIM / COL_CHUNKS) + 2 * khi;

    v2f bones = {1.0f, 1.0f};                              // ones: layout-invariant
    v8f c = {0.f, 0.f, 0.f, 0.f, 0.f, 0.f, 0.f, 0.f};

    #pragma unroll 8
    for (int s = 0; s < (N_DIM / COL_CHUNKS) / 4; ++s) {   // 64 steps
        v2f a = *(const v2f*)(p + 4 * s);
        // 8 args: (neg_a, A, neg_b, B, c_mod, C, reuse_a, reuse_b)
        c = __builtin_amdgcn_wmma_f32_16x16x4_f32(false, a, false, bones,
                                                  (short)0, c, false, false);
    }

    // Column N=0: lane 0 holds rows 0..7 in c[0..7], lane 16 rows 8..15.
    if ((lane & 15) == 0) {
        const int half = lane >> 4;
        float* dst = partial + (size_t)chunk * BN
                   + (size_t)batch * N_DIM + row_base + half * 8;
        #pragma unroll
        for (int r = 0; r < 8; ++r) dst[r] = c[r];
    }
}

// ---------------------------------------------------------------------------
// Pass 1b: deterministic fixed-order reduction of the 8 partials + rsqrt.
// 32768 rows -> 128 blocks. Negligible traffic (1.125 MB).
// ---------------------------------------------------------------------------
__global__ __launch_bounds__(256)
void laplacian_degree_finalize(const float* __restrict__ partial,
                               float* __restrict__ dinv) {
    const int idx = blockIdx.x * 256 + threadIdx.x;        // < BN
    float s = 0.0f;
    #pragma unroll
    for (int cth = 0; cth < COL_CHUNKS; ++cth) s += partial[(size_t)cth * BN + idx];
    dinv[idx] = rsqrtf(s);
}

// ---------------------------------------------------------------------------
// Pass 2: out[b,i,j] = (i==j) - dinv[b,i] * A[b,i,j] * dinv[b,j].
// One float4 per thread. Non-temporal loads/stores for the streamed 512 MB;
// blocks iterate in REVERSE linear order so the tail of A (still resident in
// the 192 MB L2 from pass 1) is consumed first.
// ---------------------------------------------------------------------------
__global__ __launch_bounds__(256)
void laplacian_scale(const float* __restrict__ A, const float* __restrict__ dinv,
                     float* __restrict__ out) {
    const size_t per_batch4 = (size_t)N_DIM * (N_DIM / 4);
    const size_t idx = (size_t)(gridDim.x - 1u - blockIdx.x) * blockDim.x + threadIdx.x;

    const int    b   = (int)(idx / per_batch4);
    const size_t rem = idx % per_batch4;
    const int    i   = (int)(rem / (N_DIM / 4));
    const int    jc  = (int)(rem % (N_DIM / 4)) * 4;

    const v4f a  = __builtin_nontemporal_load((const v4f*)(A + idx * 4));
    const float di = dinv[(size_t)b * N_DIM + i];
    const v4f dj = *(const v4f*)(dinv + (size_t)b * N_DIM + jc);   // 16B aligned

    v4f r;
    r.x = -(di * dj.x) * a.x;
    r.y = -(di * dj.y) * a.y;
    r.z = -(di * dj.z) * a.z;
    r.w = -(di * dj.w) * a.w;

    const int t = i - jc;            // diagonal lands in this float4?
    if ((unsigned)t < 4u) r[t] += 1.0f;

    __builtin_nontemporal_store(r, (v4f*)(out + idx * 4));
}

// ---------------------------------------------------------------------------
extern "C" void kernel_launch(void* const* d_in, const int* in_sizes, int n_in,
                              void* d_out, int out_size, void* d_ws, size_t ws_size,
                              hipStream_t stream) {
    const float* A   = (const float*)d_in[0];
    float*       out = (float*)d_out;
    float* partial = (float*)d_ws;                      // 8 * 32768 f32 = 1 MB
    float* dinv    = partial + (size_t)COL_CHUNKS * BN; // 32768 f32 = 128 KB

    dim3 blk(256);

    // Pass 1a: 16 batches * 16 strips * 8 column chunks = 2048 blocks.
    dim3 grid1(B_DIM * (N_DIM / 128) * COL_CHUNKS);
    laplacian_degree_partial<<<grid1, blk, 0, stream>>>(A, partial);

    // Pass 1b: 32768 rows / 256 = 128 blocks.
    dim3 grid1b(BN / 256);
    laplacian_degree_finalize<<<grid1b, blk, 0, stream>>>(partial, dinv);

    // Pass 2: 16*2048*2048/4 float4s / 256 threads = 65536 blocks.
    const size_t total4 = (size_t)B_DIM * N_DIM * (N_DIM / 4);
    dim3 grid2((unsigned)(total4 / 256));
    laplacian_scale<<<grid2, blk, 0, stream>>>(A, dinv, out);
}